// DynamicScatter_23845658427908
// MI455X (gfx1250) — compile-verified
//
#include <hip/hip_runtime.h>
#include <stdint.h>

typedef unsigned int u32;
typedef int v4i __attribute__((vector_size(16)));  // matches builtin's int __vector(4)

#define GRID_DIM 64
#define NBINS (GRID_DIM * GRID_DIM * GRID_DIM) /* 262144 voxel bins */
#define GLOBAL_AS __attribute__((address_space(1)))
#define LDS_AS __attribute__((address_space(3)))

// ---------------------------------------------------------------------------
// gfx1250 async global->LDS copy (CDNA5 path). Tracked by ASYNCcnt; fall back
// to a plain 128-bit copy if the builtin is unavailable on this toolchain.
// Builtin signature (from hipcc diagnostic): param0 = v4i in AS1 (__device__),
// so cast both sides to vector-typed, address-space-qualified pointers.
// ---------------------------------------------------------------------------
__device__ __forceinline__ void async_copy_b128(const u32* g, u32* l) {
#if defined(__gfx1250__) && __has_builtin(__builtin_amdgcn_global_load_async_to_lds_b128)
  __builtin_amdgcn_global_load_async_to_lds_b128(
      (GLOBAL_AS v4i*)(void*)(u32*)g,
      (LDS_AS v4i*)(void*)l, 0, 0);
#else
  *(uint4*)l = *(const uint4*)g;
#endif
}

__device__ __forceinline__ void wait_async0() {
#if defined(__gfx1250__) && __has_builtin(__builtin_amdgcn_s_wait_asynccnt)
  __builtin_amdgcn_s_wait_asynccnt(0);
#elif defined(__gfx1250__)
  asm volatile("s_wait_asynccnt 0" ::: "memory");
#endif
}

// ---------------------------------------------------------------------------
// memset32: vectorized 128-bit stores; used to zero feats / set coors to -1 /
// zero the count+cursor tables. ~268MB total => ~12us at 23.3 TB/s.
// ---------------------------------------------------------------------------
__global__ void memset32_kernel(u32* __restrict__ dst, u32 val, long long nd) {
  long long i = ((long long)blockIdx.x * blockDim.x + threadIdx.x) * 4;
  if (i + 4 <= nd) {
    *(uint4*)(dst + i) = make_uint4(val, val, val, val);
  } else {
    for (int k = 0; k < 4; ++k)
      if (i + k < nd) dst[i + k] = val;
  }
}

// ---------------------------------------------------------------------------
// Pass 1: per-voxel point counts. coors tile (256 pts * 3 dwords) is staged
// into LDS with async b128 loads, then each point does one u32 atomic.
// ---------------------------------------------------------------------------
__global__ void count_kernel(const int* __restrict__ coors,
                             u32* __restrict__ counts, int n) {
  __shared__ u32 lds[256 * 3];
  const int t = threadIdx.x;
  const long long base = (long long)blockIdx.x * 256;
  const long long dwBase = base * 3;
  const long long totalDw = (long long)n * 3;
  long long rem = totalDw - dwBase;
  int dwCount = rem > 768 ? 768 : (int)rem;
  int nLoads = (dwCount + 3) >> 2;
  if (t < nLoads) async_copy_b128((const u32*)coors + dwBase + t * 4, &lds[t * 4]);
  wait_async0();
  __syncthreads();
  long long i = base + t;
  if (i < n) {
    u32 key = (lds[t * 3 + 0] << 12) | (lds[t * 3 + 1] << 6) | lds[t * 3 + 2];
    atomicAdd(&counts[key], 1u);
  }
}

// ---------------------------------------------------------------------------
// Block-wide exclusive scan (Hillis-Steele over 256 threads in LDS).
// ---------------------------------------------------------------------------
__device__ __forceinline__ u32 block_exscan256(u32 v, u32* sm, u32* total) {
  const int t = threadIdx.x;
  sm[t] = v;
  __syncthreads();
  for (int off = 1; off < 256; off <<= 1) {
    u32 x = (t >= off) ? sm[t - off] : 0u;
    __syncthreads();
    sm[t] += x;
    __syncthreads();
  }
  u32 incl = sm[t];
  u32 tot = sm[255];
  __syncthreads();  // allow caller to reuse sm
  if (total) *total = tot;
  return incl - v;
}

// Pass 2a: per-block totals of (occupancy, counts). 256 blocks * 1024 bins.
__global__ void scan_partials_kernel(const u32* __restrict__ counts,
                                     u32* __restrict__ blkOcc,
                                     u32* __restrict__ blkCnt) {
  __shared__ u32 sm[256];
  const int t = threadIdx.x;
  const int b0 = (blockIdx.x * 256 + t) * 4;
  u32 occ = 0, cs = 0;
  for (int k = 0; k < 4; ++k) {
    u32 c = counts[b0 + k];
    occ += (c > 0u);
    cs += c;
  }
  u32 to, tc;
  (void)block_exscan256(occ, sm, &to);
  (void)block_exscan256(cs, sm, &tc);
  if (t == 0) { blkOcc[blockIdx.x] = to; blkCnt[blockIdx.x] = tc; }
}

// Pass 2b: single-block scan of the 256 block totals; writes numOccupied.
__global__ void scan_offsets_kernel(u32* __restrict__ blkOcc,
                                    u32* __restrict__ blkCnt,
                                    u32* __restrict__ numOcc) {
  __shared__ u32 sm[256];
  const int t = threadIdx.x;
  u32 vo = blkOcc[t], vc = blkCnt[t];
  u32 to, tc;
  u32 eo = block_exscan256(vo, sm, &to);
  u32 ec = block_exscan256(vc, sm, &tc);
  blkOcc[t] = eo;
  blkCnt[t] = ec;
  if (t == 0) *numOcc = to;
}

// Pass 2c: final scan application. For each occupied bin (ascending key order
// == jnp.unique sorted order): rank -> key, list start, count; binStart for
// the scatter pass.
__global__ void scan_apply_kernel(const u32* __restrict__ counts,
                                  const u32* __restrict__ blkOcc,
                                  const u32* __restrict__ blkCnt,
                                  u32* __restrict__ binStart,
                                  u32* __restrict__ keys,
                                  u32* __restrict__ rowStart,
                                  u32* __restrict__ cntOf) {
  __shared__ u32 sm[256];
  const int t = threadIdx.x;
  const int b0 = (blockIdx.x * 256 + t) * 4;
  u32 c[4];
  u32 occ = 0, cs = 0;
  for (int k = 0; k < 4; ++k) {
    c[k] = counts[b0 + k];
    occ += (c[k] > 0u);
    cs += c[k];
  }
  u32 occOff = block_exscan256(occ, sm, nullptr) + blkOcc[blockIdx.x];
  u32 cntOff = block_exscan256(cs, sm, nullptr) + blkCnt[blockIdx.x];
  for (int k = 0; k < 4; ++k) {
    binStart[b0 + k] = cntOff;
    if (c[k] > 0u) {
      keys[occOff] = (u32)(b0 + k);
      rowStart[occOff] = cntOff;
      cntOf[occOff] = c[k];
      occOff++;
    }
    cntOff += c[k];
  }
}

// ---------------------------------------------------------------------------
// Pass 3: scatter point indices into contiguous per-voxel lists.
// ---------------------------------------------------------------------------
__global__ void fill_kernel(const int* __restrict__ coors,
                            const u32* __restrict__ binStart,
                            u32* __restrict__ cursor,
                            u32* __restrict__ ptIdx, int n) {
  __shared__ u32 lds[256 * 3];
  const int t = threadIdx.x;
  const long long base = (long long)blockIdx.x * 256;
  const long long dwBase = base * 3;
  const long long totalDw = (long long)n * 3;
  long long rem = totalDw - dwBase;
  int dwCount = rem > 768 ? 768 : (int)rem;
  int nLoads = (dwCount + 3) >> 2;
  if (t < nLoads) async_copy_b128((const u32*)coors + dwBase + t * 4, &lds[t * 4]);
  wait_async0();
  __syncthreads();
  long long i = base + t;
  if (i < n) {
    u32 key = (lds[t * 3 + 0] << 12) | (lds[t * 3 + 1] << 6) | lds[t * 3 + 2];
    u32 slot = atomicAdd(&cursor[key], 1u);
    ptIdx[binStart[key] + slot] = (u32)i;
  }
}

// ---------------------------------------------------------------------------
// Pass 4: gather-mean. One 64-lane group per output row; lane l owns feature
// l so each point row read is a fully coalesced 256B burst. Small lists are
// sorted in LDS (deterministic summation order across replays).
// ---------------------------------------------------------------------------
__global__ void gather_kernel(const float* __restrict__ points,
                              const u32* __restrict__ keys,
                              const u32* __restrict__ rowStart,
                              const u32* __restrict__ cntOf,
                              const u32* __restrict__ ptIdx,
                              const u32* __restrict__ numOccPtr,
                              float* __restrict__ feats,
                              int* __restrict__ vcoors) {
  __shared__ u32 list[4][128];
  const u32 numOcc = *numOccPtr;
  const int g = threadIdx.x >> 6;   // row group within block (4 per block)
  const int l = threadIdx.x & 63;   // feature lane
  const u32 row = blockIdx.x * 4 + g;
  const bool active = row < numOcc;
  u32 cnt = 0, start = 0, key = 0;
  if (active) { cnt = cntOf[row]; start = rowStart[row]; key = keys[row]; }
  u32 mn = cnt < 128u ? cnt : 128u;
  for (u32 p = l; p < mn; p += 64) list[g][p] = ptIdx[start + p];
  __syncthreads();
  if (active && l == 0 && cnt <= 128u) {  // insertion sort (avg cnt ~ 4)
    for (int a = 1; a < (int)cnt; ++a) {
      u32 v = list[g][a];
      int b = a - 1;
      while (b >= 0 && list[g][b] > v) { list[g][b + 1] = list[g][b]; --b; }
      list[g][b + 1] = v;
    }
  }
  __syncthreads();
  if (active) {
    float s = 0.0f;
    if (cnt <= 128u) {
      for (u32 p = 0; p < cnt; ++p) {
        u32 idx = list[g][p];
        if (p + 1 < cnt)
          __builtin_prefetch(&points[(size_t)list[g][p + 1] * 64], 0, 0);
        s += points[(size_t)idx * 64 + l];
      }
    } else {  // robustness fallback for pathological voxels
      for (u32 p = 0; p < cnt; ++p) s += points[(size_t)ptIdx[start + p] * 64 + l];
    }
    feats[(size_t)row * 64 + l] = s / (float)cnt;
    if (l < 3) {
      int v = (l == 0) ? (int)(key >> 12)
            : (l == 1) ? (int)((key >> 6) & 63)
                       : (int)(key & 63);
      vcoors[(size_t)row * 3 + l] = v;
    }
  }
}

// ---------------------------------------------------------------------------
extern "C" void kernel_launch(void* const* d_in, const int* in_sizes, int n_in,
                              void* d_out, int out_size, void* d_ws, size_t ws_size,
                              hipStream_t stream) {
  const float* points = (const float*)d_in[0];
  const int* coors = (const int*)d_in[1];
  const int n = in_sizes[1] / 3;  // 1,000,000

  float* feats = (float*)d_out;                   // [n,64] f32
  int* vcoors = (int*)d_out + (size_t)n * 64;     // [n,3] i32 (bit-identical 4B)

  // Workspace layout (u32 units). ~10.3 MB total.
  u32* ws = (u32*)d_ws;
  u32* counts   = ws;                    // NBINS
  u32* cursor   = counts + NBINS;        // NBINS (contiguous with counts -> one memset)
  u32* binStart = cursor + NBINS;        // NBINS
  u32* keys     = binStart + NBINS;      // NBINS
  u32* rowStart = keys + NBINS;          // NBINS
  u32* cntOf    = rowStart + NBINS;      // NBINS
  u32* blkOcc   = cntOf + NBINS;         // 256
  u32* blkCnt   = blkOcc + 256;          // 256
  u32* numOcc   = blkCnt + 256;          // 1 (+pad)
  u32* ptIdx    = numOcc + 64;           // n

  const int BS = 256;
  auto blocks_for_dw = [](long long nd) -> u32 {
    return (u32)(((nd + 3) / 4 + 255) / 256);
  };

  // Init: zero feats, -1 coors, zero counts+cursor (d_out/d_ws are poisoned).
  long long featsDw = (long long)n * 64;
  memset32_kernel<<<blocks_for_dw(featsDw), BS, 0, stream>>>((u32*)feats, 0u, featsDw);
  memset32_kernel<<<blocks_for_dw((long long)n * 3), BS, 0, stream>>>(
      (u32*)vcoors, 0xFFFFFFFFu, (long long)n * 3);
  memset32_kernel<<<blocks_for_dw(2LL * NBINS), BS, 0, stream>>>(counts, 0u, 2LL * NBINS);

  const u32 ptBlocks = (u32)((n + 255) / 256);
  count_kernel<<<ptBlocks, BS, 0, stream>>>(coors, counts, n);
  scan_partials_kernel<<<256, BS, 0, stream>>>(counts, blkOcc, blkCnt);
  scan_offsets_kernel<<<1, BS, 0, stream>>>(blkOcc, blkCnt, numOcc);
  scan_apply_kernel<<<256, BS, 0, stream>>>(counts, blkOcc, blkCnt,
                                            binStart, keys, rowStart, cntOf);
  fill_kernel<<<ptBlocks, BS, 0, stream>>>(coors, binStart, cursor, ptIdx, n);
  gather_kernel<<<NBINS / 4, BS, 0, stream>>>(points, keys, rowStart, cntOf,
                                              ptIdx, numOcc, feats, vcoors);
}